// GCNeXt_11811160064213
// MI455X (gfx1250) — compile-verified
//
#include <hip/hip_runtime.h>

// GCNeXt block on MI455X (gfx1250, wave32, WMMA + Tensor Data Mover).
//
// Roofline: ~29 GFLOP total, dominated by the N x N x C distance GEMM (17.2 GF).
// x (16.8 MB) is L2-resident (192 MB L2), so the binding resource is L2 bandwidth.
// KNN blocking: 128-row blocks, register-resident A strips (128 VGPRs/lane),
// B tiles staged to LDS by TDM (tensor_load_to_lds, TENSORcnt double-buffering)
// -> L2 reads drop from ~4 GB to ~270 MB. All GEMMs use V_WMMA_F32_16X16X4_F32
// (fp32-exact vs. the reference; critical for top-k index ordering).

#define BB 8
#define CC 256
#define NN 2048
#define KK 3
#define WW 128

typedef float v2f __attribute__((ext_vector_type(2)));
typedef float v8f __attribute__((ext_vector_type(8)));
typedef unsigned int u32x4 __attribute__((ext_vector_type(4)));
typedef int i32x4 __attribute__((ext_vector_type(4)));
typedef int i32x8 __attribute__((ext_vector_type(8)));

// ---------------------------------------------------------------- WMMA helper
// D = A(16x4) * B(4x16) + C(16x16), fp32.
// A: lane l holds M=l&15; VGPR0/1 = K{0,1} (lanes 0-15) / K{2,3} (lanes 16-31).
// B (mirror): lane l holds N=l&15; same K split by half-wave.
// C/D: lane l holds N=l&15; VGPR v holds M = v + 8*(l>>4).
__device__ inline v8f wmma4(v2f a, v2f b, v8f c) {
#if __has_builtin(__builtin_amdgcn_wmma_f32_16x16x4_f32)
  return __builtin_amdgcn_wmma_f32_16x16x4_f32(false, a, false, b, (short)0, c,
                                               false, false);
#else
  int l = (int)(threadIdx.x & 31);
  int n = l & 15;
  float b0 = __shfl(b.x, n, 32), b1 = __shfl(b.y, n, 32);
  float b2 = __shfl(b.x, n + 16, 32), b3 = __shfl(b.y, n + 16, 32);
  int mb = (l >> 4) * 8;
#pragma unroll
  for (int v = 0; v < 8; ++v) {
    int m = mb + v;
    float a0 = __shfl(a.x, m, 32), a1 = __shfl(a.y, m, 32);
    float a2 = __shfl(a.x, m + 16, 32), a3 = __shfl(a.y, m + 16, 32);
    c[v] += a0 * b0 + a1 * b1 + a2 * b2 + a3 * b3;
  }
  return c;
#endif
}

// ------------------------------------------------------- TDM staging helpers
#define HAS_TDM __has_builtin(__builtin_amdgcn_tensor_load_to_lds)

__device__ __forceinline__ unsigned lds_off(const void* p) {
  // Flat LDS aperture: addr[31:0] is the DS-space byte address.
  return (unsigned)(unsigned long long)(uintptr_t)p;
}

#if HAS_TDM
// 2-D tile load, data_size = 4B. D# bitfields per CDNA5 ISA.
__device__ __forceinline__ void tdm_load_2d(const void* gaddr, unsigned lds_addr,
                                            unsigned tile_d0, unsigned tile_d1,
                                            unsigned tens_d0, unsigned tens_d1,
                                            unsigned stride0) {
  unsigned long long ga = (unsigned long long)(uintptr_t)gaddr;
  u32x4 g0;
  g0[0] = 1u;                                            // count=1 (valid user D#)
  g0[1] = lds_addr;                                      // lds_addr (bytes)
  g0[2] = (unsigned)(ga & 0xFFFFFFFFu);                  // global_addr[31:0]
  g0[3] = (unsigned)((ga >> 32) & 0x01FFFFFFu) | (2u << 30);  // addr[56:32], type=2
  i32x8 g1;
  g1[0] = (int)(2u << 16);                               // data_size=4B; mask=0
  g1[1] = (int)((tens_d0 & 0xFFFFu) << 16);              // tensor_dim0[15:0]
  g1[2] = (int)(((tens_d0 >> 16) & 0xFFFFu) | ((tens_d1 & 0xFFFFu) << 16));
  g1[3] = (int)(((tens_d1 >> 16) & 0xFFFFu) | ((tile_d0 & 0xFFFFu) << 16));
  g1[4] = (int)(tile_d1 & 0xFFFFu);                      // tile_dim1; tile_dim2=0
  g1[5] = (int)stride0;                                  // tensor_dim0_stride[31:0]
  g1[6] = 0;
  g1[7] = 0;
  i32x4 z4 = {0, 0, 0, 0};
#if __clang_major__ >= 23
  i32x8 z8 = {0, 0, 0, 0, 0, 0, 0, 0};
  __builtin_amdgcn_tensor_load_to_lds(g0, g1, z4, z4, z8, 0);
#else
  __builtin_amdgcn_tensor_load_to_lds(g0, g1, z4, z4, 0);
#endif
}
#endif

// Stage a d1 x d0 tile (row stride `stride0` elements) of f32 into LDS.
__device__ __forceinline__ void stage2d(float* dst, const float* src, int d0,
                                        int d1, int stride0, int tid) {
#if HAS_TDM
  if ((tid >> 5) == 0)  // one wave issues; TDM ignores EXEC, one DMA per wave
    tdm_load_2d(src, lds_off(dst), (unsigned)d0, (unsigned)d1,
                (unsigned)stride0, (unsigned)d1, (unsigned)stride0);
#else
  int n = d0 * d1;
  for (int i = tid; i < n; i += (int)blockDim.x)
    dst[i] = src[(size_t)(i / d0) * stride0 + (i % d0)];
#endif
}

__device__ __forceinline__ void stage_wait(int tid) {
#if HAS_TDM
  if ((tid >> 5) == 0) {
#if __has_builtin(__builtin_amdgcn_s_wait_tensorcnt)
    __builtin_amdgcn_s_wait_tensorcnt(0);
#else
    asm volatile("s_wait_tensorcnt 0x0" ::: "memory");
#endif
  }
#endif
}

// ---------------------------------------------------------------- xx = sum(x*x)
__global__ void xx_kernel(const float* __restrict__ x, float* __restrict__ xx) {
  int id = blockIdx.x * blockDim.x + threadIdx.x;  // B*N
  if (id >= BB * NN) return;
  int b = id / NN, n = id % NN;
  const float* xb = x + (size_t)b * CC * NN + n;
  float s = 0.f;
  for (int c = 0; c < CC; ++c) {
    float v = xb[(size_t)c * NN];
    s += v * v;
  }
  xx[id] = s;
}

// ---------------------------------------------- fused X^T X + top-3 neighbors
// 128-row block (8 waves x 16 rows). A strips register-resident; B tiles
// TDM-staged into LDS with double buffering; row norms TDM-staged once.
__global__ void __launch_bounds__(256) knn_kernel(const float* __restrict__ x,
                                                  const float* __restrict__ xx,
                                                  int* __restrict__ idx) {
  int b = blockIdx.x >> 4;              // N/128 = 16 strips per batch
  int m0 = (blockIdx.x & 15) * 128;
  int tid = threadIdx.x, wave = tid >> 5, l = tid & 31;
  int ncol = l & 15, kb = (l >> 4) * 2, mbase = (l >> 4) * 8;
  const float* xb = x + (size_t)b * CC * NN;

  __shared__ float Bbuf[2][CC * 16];    // 2 x 16 KB column tiles
  __shared__ float xxs[NN];             // 8 KB row norms
  __shared__ float scan[8][16][17];     // per-wave pd tile scratch

  stage2d(xxs, xx + (size_t)b * NN, NN, 1, NN, tid);
  stage2d(Bbuf[0], xb, 16, CC, NN, tid);

  // A strip: 16 rows x 256 channels for this wave, kept in 128 VGPRs/lane.
  int mrow = m0 + wave * 16 + ncol;
  v2f areg[CC / 4];
#pragma unroll
  for (int s = 0; s < CC / 4; ++s) {
    areg[s].x = xb[(size_t)(4 * s + kb) * NN + mrow];
    areg[s].y = xb[(size_t)(4 * s + kb + 1) * NN + mrow];
  }

  stage_wait(tid);
  __syncthreads();

  float xxrow[8];
#pragma unroll
  for (int v = 0; v < 8; ++v) xxrow[v] = xxs[m0 + wave * 16 + mbase + v];

  float bv0 = -3.4e38f, bv1 = -3.4e38f, bv2 = -3.4e38f;
  int bi0 = 0, bi1 = 0, bi2 = 0;

  for (int t = 0; t < NN / 16; ++t) {
    int cur = t & 1;
    if (t + 1 < NN / 16)  // prefetch next tile into the other buffer
      stage2d(Bbuf[cur ^ 1], xb + (size_t)(t + 1) * 16, 16, CC, NN, tid);

    const float* Bt = Bbuf[cur];
    v8f acc = {};
#pragma unroll
    for (int s = 0; s < CC / 4; ++s) {
      v2f bf;
      bf.x = Bt[(4 * s + kb) * 16 + ncol];
      bf.y = Bt[(4 * s + kb + 1) * 16 + ncol];
      acc = wmma4(areg[s], bf, acc);
    }
    float xxn = xxs[t * 16 + ncol];
#pragma unroll
    for (int v = 0; v < 8; ++v)
      scan[wave][mbase + v][ncol] = 2.f * acc[v] - xxrow[v] - xxn;  // -||xm-xn||^2
    __syncthreads();  // publish pd tile; retire reads of Bt

    if (l < 16) {
      for (int j = 0; j < 16; ++j) {
        float pv = scan[wave][l][j];
        int gi = t * 16 + j;
        if (pv > bv0) {
          bv2 = bv1; bi2 = bi1; bv1 = bv0; bi1 = bi0; bv0 = pv; bi0 = gi;
        } else if (pv > bv1) {
          bv2 = bv1; bi2 = bi1; bv1 = pv; bi1 = gi;
        } else if (pv > bv2) {
          bv2 = pv; bi2 = gi;
        }
      }
    }
    stage_wait(tid);   // next tile landed (wave0 TENSORcnt)
    __syncthreads();   // publish next tile; retire scan reads
  }
  if (l < 16) {
    int* o = idx + ((size_t)b * NN + m0 + wave * 16 + l) * KK;
    o[0] = bi0; o[1] = bi1; o[2] = bi2;
  }
}

// -------------------------- t1: 1x1 conv = W x C GEMM + bias + relu (TDM B tile)
__global__ void __launch_bounds__(256) t1_kernel(const float* __restrict__ x,
                                                 const float* __restrict__ w,
                                                 const float* __restrict__ bias,
                                                 float* __restrict__ h1) {
  __shared__ float Bt[CC * 16];
  int b = blockIdx.x >> 7;
  int n0 = (blockIdx.x & 127) * 16;
  int tid = threadIdx.x, wave = tid >> 5, l = tid & 31;
  int ncol = l & 15, kb = (l >> 4) * 2, mbase = (l >> 4) * 8;
  const float* xb = x + (size_t)b * CC * NN;

  stage2d(Bt, xb + n0, 16, CC, NN, tid);
  stage_wait(tid);
  __syncthreads();

  int m0 = wave * 16;  // 8 waves cover W=128
  v8f acc = {};
#pragma unroll
  for (int k = 0; k < CC; k += 4) {
    v2f a = *(const v2f*)&w[(size_t)(m0 + ncol) * CC + k + kb];
    v2f bf;
    bf.x = Bt[(k + kb) * 16 + ncol];
    bf.y = Bt[(k + kb + 1) * 16 + ncol];
    acc = wmma4(a, bf, acc);
  }
  float* hb = h1 + (size_t)b * WW * NN;
#pragma unroll
  for (int v = 0; v < 8; ++v) {
    int m = m0 + mbase + v;
    float r = acc[v] + bias[m];
    hb[(size_t)m * NN + n0 + ncol] = r > 0.f ? r : 0.f;
  }
}

// ------------------- t2: grouped temporal conv, kernel 3, pad 1 (tiny; VALU)
__global__ void t2_kernel(const float* __restrict__ h1, const float* __restrict__ w,
                          const float* __restrict__ bias, float* __restrict__ h2) {
  int id = blockIdx.x * blockDim.x + threadIdx.x;  // B*W*N
  if (id >= BB * WW * NN) return;
  int n = id % NN;
  int bw = id / NN;
  int wch = bw % WW;
  int b = bw / WW;
  int gbase = (wch >> 2) << 2;
  const float* hb = h1 + (size_t)b * WW * NN;
  float s = bias[wch];
#pragma unroll
  for (int i = 0; i < 4; ++i) {
    const float* row = hb + (size_t)(gbase + i) * NN;
    const float* wr = w + (size_t)(wch * 4 + i) * 3;
    if (n > 0) s += wr[0] * row[n - 1];
    s += wr[1] * row[n];
    if (n < NN - 1) s += wr[2] * row[n + 1];
  }
  h2[id] = s > 0.f ? s : 0.f;
}

// ------- fused semantic branch: s1 (edge GEMM) -> s2 -> s3 -> max over k
__global__ void __launch_bounds__(256) sem_kernel(
    const float* __restrict__ x, const int* __restrict__ idx,
    const float* __restrict__ s1w, const float* __restrict__ s1b,
    const float* __restrict__ s2w, const float* __restrict__ s2b,
    const float* __restrict__ s3w, const float* __restrict__ s3b,
    float* __restrict__ sout) {
  int b = blockIdx.x >> 7;
  int n0 = (blockIdx.x & 127) * 16;
  int tid = threadIdx.x, wave = tid >> 5, l = tid & 31;
  int ncol = l & 15, kb = (l >> 4) * 2, mbase = (l >> 4) * 8;
  const float* xb = x + (size_t)b * CC * NN;

  __shared__ int nbr[KK][16];
  __shared__ float smem[2 * WW * 48];  // g1 | g2; center tile aliases g2
  float* g1 = smem;                    // [WW][48] s1 output (relu)
  float* g2 = smem + WW * 48;          // [WW][48] s2 output (relu)
  float* Ct = g2;                      // [CC][16] TDM-staged center features

  stage2d(Ct, xb + n0, 16, CC, NN, tid);  // x[:, n0:n0+16]
  if (tid < 16) {
    const int* ip = idx + ((size_t)b * NN + n0 + tid) * KK;
    nbr[0][tid] = ip[0]; nbr[1][tid] = ip[1]; nbr[2][tid] = ip[2];
  }
  stage_wait(tid);
  __syncthreads();

  // s1: W x 2C GEMM. Neighbor half gathered from global (L2-hot), center
  // half from the TDM-staged LDS tile.
  {
    int m0 = wave * 16;  // 8 waves cover W=128
    for (int ct = 0; ct < KK; ++ct) {
      int pnbr = nbr[ct][ncol];
      v8f acc = {};
#pragma unroll
      for (int k = 0; k < 2 * CC; k += 4) {
        v2f a = *(const v2f*)&s1w[(size_t)(m0 + ncol) * (2 * CC) + k + kb];
        int ch = k + kb;  // even: ch, ch+1 stay on one side of the concat
        v2f bf;
        if (ch < CC) {
          const float* src = xb + (size_t)ch * NN;
          bf.x = src[pnbr];
          bf.y = src[NN + pnbr];
        } else {
          bf.x = Ct[(ch - CC) * 16 + ncol];
          bf.y = Ct[(ch - CC + 1) * 16 + ncol];
        }
        acc = wmma4(a, bf, acc);
      }
#pragma unroll
      for (int v = 0; v < 8; ++v) {
        int m = m0 + mbase + v;
        float r = acc[v] + s1b[m];
        g1[m * 48 + ct * 16 + ncol] = r > 0.f ? r : 0.f;
      }
    }
  }
  __syncthreads();

  // s2: grouped 1x1 (4 in-channels per group), LDS -> LDS (overwrites Ct).
  for (int e = tid; e < WW * 48; e += 256) {
    int wch = e / 48, col = e % 48;
    int gb = (wch >> 2) << 2;
    float s = s2b[wch];
#pragma unroll
    for (int i = 0; i < 4; ++i) s += s2w[wch * 4 + i] * g1[(gb + i) * 48 + col];
    g2[wch * 48 + col] = s > 0.f ? s : 0.f;
  }
  __syncthreads();

  // s3: C x W GEMM from LDS, max over the 3 neighbor column tiles + bias.
  float* sb = sout + (size_t)b * CC * NN;
  for (int half = 0; half < 2; ++half) {
    int m0 = half * 128 + wave * 16;  // 8 waves x 2 halves cover C=256
    v8f acc0 = {}, acc1 = {}, acc2 = {};
#pragma unroll
    for (int k = 0; k < WW; k += 4) {
      v2f a = *(const v2f*)&s3w[(size_t)(m0 + ncol) * WW + k + kb];
      v2f b0, b1, b2;
      b0.x = g2[(k + kb) * 48 + ncol];      b0.y = g2[(k + kb + 1) * 48 + ncol];
      b1.x = g2[(k + kb) * 48 + 16 + ncol]; b1.y = g2[(k + kb + 1) * 48 + 16 + ncol];
      b2.x = g2[(k + kb) * 48 + 32 + ncol]; b2.y = g2[(k + kb + 1) * 48 + 32 + ncol];
      acc0 = wmma4(a, b0, acc0);
      acc1 = wmma4(a, b1, acc1);
      acc2 = wmma4(a, b2, acc2);
    }
#pragma unroll
    for (int v = 0; v < 8; ++v) {
      int m = m0 + mbase + v;
      float r = fmaxf(acc0[v], fmaxf(acc1[v], acc2[v])) + s3b[m];
      sb[(size_t)m * NN + n0 + ncol] = r;
    }
  }
}

// ---- t3: C x W GEMM (TDM-staged B) fused with relu(tout + identity + sout)
__global__ void __launch_bounds__(256) t3_kernel(const float* __restrict__ h2,
                                                 const float* __restrict__ w,
                                                 const float* __restrict__ bias,
                                                 const float* __restrict__ x,
                                                 const float* __restrict__ sout,
                                                 float* __restrict__ out) {
  __shared__ float Ht[WW * 16];
  int b = blockIdx.x >> 7;
  int n0 = (blockIdx.x & 127) * 16;
  int tid = threadIdx.x, wave = tid >> 5, l = tid & 31;
  int ncol = l & 15, kb = (l >> 4) * 2, mbase = (l >> 4) * 8;
  const float* hb = h2 + (size_t)b * WW * NN;

  stage2d(Ht, hb + n0, 16, WW, NN, tid);
  stage_wait(tid);
  __syncthreads();

  for (int half = 0; half < 2; ++half) {
    int m0 = half * 128 + wave * 16;
    v8f acc = {};
#pragma unroll
    for (int k = 0; k < WW; k += 4) {
      v2f a = *(const v2f*)&w[(size_t)(m0 + ncol) * WW + k + kb];
      v2f bf;
      bf.x = Ht[(k + kb) * 16 + ncol];
      bf.y = Ht[(k + kb + 1) * 16 + ncol];
      acc = wmma4(a, bf, acc);
    }
#pragma unroll
    for (int v = 0; v < 8; ++v) {
      int m = m0 + mbase + v;
      size_t off = ((size_t)b * CC + m) * NN + n0 + ncol;
      float r = acc[v] + bias[m] + x[off] + sout[off];
      out[off] = r > 0.f ? r : 0.f;
    }
  }
}

extern "C" void kernel_launch(void* const* d_in, const int* in_sizes, int n_in,
                              void* d_out, int out_size, void* d_ws, size_t ws_size,
                              hipStream_t stream) {
  (void)in_sizes; (void)n_in; (void)out_size; (void)ws_size;
  const float* x   = (const float*)d_in[0];
  const float* t1w = (const float*)d_in[1];
  const float* t1b = (const float*)d_in[2];
  const float* t2w = (const float*)d_in[3];
  const float* t2b = (const float*)d_in[4];
  const float* t3w = (const float*)d_in[5];
  const float* t3b = (const float*)d_in[6];
  const float* s1w = (const float*)d_in[7];
  const float* s1b = (const float*)d_in[8];
  const float* s2w = (const float*)d_in[9];
  const float* s2b = (const float*)d_in[10];
  const float* s3w = (const float*)d_in[11];
  const float* s3b = (const float*)d_in[12];
  float* out = (float*)d_out;

  float* xx   = (float*)d_ws;                        // B*N
  int*   idx  = (int*)(xx + (size_t)BB * NN);        // B*N*K
  float* h1   = (float*)(idx + (size_t)BB * NN * KK);
  float* h2   = h1 + (size_t)BB * WW * NN;
  float* sout = h2 + (size_t)BB * WW * NN;           // B*C*N

  xx_kernel<<<(BB * NN) / 256, 256, 0, stream>>>(x, xx);
  knn_kernel<<<BB * (NN / 128), 256, 0, stream>>>(x, xx, idx);
  t1_kernel<<<BB * (NN / 16), 256, 0, stream>>>(x, t1w, t1b, h1);
  t2_kernel<<<(BB * WW * NN) / 256, 256, 0, stream>>>(h1, t2w, t2b, h2);
  sem_kernel<<<BB * (NN / 16), 256, 0, stream>>>(x, idx, s1w, s1b, s2w, s2b,
                                                 s3w, s3b, sout);
  t3_kernel<<<BB * (NN / 16), 256, 0, stream>>>(h2, t3w, t3b, x, sout, out);
}